// VectorQuantizer_52140902973506
// MI455X (gfx1250) — compile-verified
//
#include <hip/hip_runtime.h>

typedef float v2f __attribute__((ext_vector_type(2)));
typedef float v4f __attribute__((ext_vector_type(4)));
typedef float v8f __attribute__((ext_vector_type(8)));

#define D_EMB 64

// ---------------------------------------------------------------------------
// Kernel 0: halfnorm[k] = 0.5*||e_k||^2 ; hist[k] = 0
// ---------------------------------------------------------------------------
__global__ __launch_bounds__(256) void vq_init_kernel(
    const float* __restrict__ emb, float* __restrict__ halfnorm,
    int* __restrict__ hist, int K) {
  int k = blockIdx.x * blockDim.x + threadIdx.x;
  if (k >= K) return;
  const float* e = emb + (size_t)k * D_EMB;
  float s = 0.0f;
#pragma unroll
  for (int j = 0; j < D_EMB; ++j) s += e[j] * e[j];
  halfnorm[k] = 0.5f * s;
  hist[k] = 0;
}

// ---------------------------------------------------------------------------
// Kernel 1: argmin over K distances per row, via fp32 WMMA 16x16x4.
// One wave handles a 16-row strip of X against all K embeddings.
// score = x.e - 0.5*||e||^2 (argmax == argmin of distance; ascending scan
// with strict > keeps jnp.argmin first-occurrence tie semantics).
// B fragments are double-buffered (b0/b1) so each tile's 16 global_load_b64
// issue as one clause and overlap the previous tile's 16 WMMAs.
// ---------------------------------------------------------------------------
__global__ __launch_bounds__(256) void vq_argmin_kernel(
    const float* __restrict__ x,        // [N, 64]
    const float* __restrict__ emb,      // [K, 64]
    const float* __restrict__ halfnorm, // [K]
    int* __restrict__ indices,          // [N]
    int* __restrict__ hist,             // [K]
    int N, int K) {
  const int wave = (blockIdx.x * blockDim.x + threadIdx.x) >> 5;
  const int lane = threadIdx.x & 31;
  const int row0 = wave * 16;
  if (row0 >= N) return;  // wave-uniform

  const int half = lane >> 4;   // 0: lanes 0-15 | 1: lanes 16-31
  const int l16  = lane & 15;

  // A fragments for the whole d=64 strip: 16 WMMA steps x v2f.
  // ISA 7.12.2 (32-bit A 16x4): M = lane&15; lanes 0-15 hold K={4s,4s+1},
  // lanes 16-31 hold K={4s+2,4s+3}.
  v2f a[16];
  const float* xrow = x + (size_t)(row0 + l16) * D_EMB;
#pragma unroll
  for (int s = 0; s < 16; ++s)
    a[s] = *(const v2f*)(xrow + 4 * s + 2 * half);

  float bestv[8];
  int   besti[8];
#pragma unroll
  for (int v = 0; v < 8; ++v) { bestv[v] = -3.4e38f; besti[v] = 0; }

  // Per-lane base for B fragments: B[k][n] = emb[col][k], col = tile*16 + l16.
  const float* ecol = emb + (size_t)l16 * D_EMB + 2 * half;
  const float* hn   = halfnorm + l16;
  const int ntiles  = K >> 4;  // 64 (even)

  auto mmtile = [&](const v2f (&b)[16]) {
    v8f acc = {};
#pragma unroll
    for (int s = 0; s < 16; ++s)
      acc = __builtin_amdgcn_wmma_f32_16x16x4_f32(
          /*neg_a=*/false, a[s], /*neg_b=*/false, b[s],
          /*c_mod=*/(short)0, acc, /*reuse_a=*/false, /*reuse_b=*/false);
    return acc;
  };
  auto update = [&](const v8f& acc, float bias, int colg) {
    // D layout: VGPR v, lanes 0-15 -> row v, lanes 16-31 -> row v+8; col=lane&15
#pragma unroll
    for (int v = 0; v < 8; ++v) {
      float t = acc[v] - bias;
      if (t > bestv[v]) { bestv[v] = t; besti[v] = colg; }
    }
  };

  v2f b0[16], b1[16];
  float bias0, bias1;

  // Prologue: tile 0 -> b0
#pragma unroll
  for (int s = 0; s < 16; ++s) b0[s] = *(const v2f*)(ecol + 4 * s);
  bias0 = hn[0];

  for (int ct = 0; ct < ntiles; ct += 2) {
    // Load tile ct+1 -> b1 (always valid: ntiles even)
    {
      const float* p = ecol + (size_t)(ct + 1) * 16 * D_EMB;
#pragma unroll
      for (int s = 0; s < 16; ++s) b1[s] = *(const v2f*)(p + 4 * s);
      bias1 = hn[(ct + 1) * 16];
    }
    // Compute tile ct with b0
    update(mmtile(b0), bias0, (ct << 4) + l16);

    // Load tile ct+2 -> b0 (clamped in-bounds on last iteration)
    {
      const int ctn  = (ct + 2 < ntiles) ? (ct + 2) : (ntiles - 2);
      const float* p = ecol + (size_t)ctn * 16 * D_EMB;
#pragma unroll
      for (int s = 0; s < 16; ++s) b0[s] = *(const v2f*)(p + 4 * s);
      bias0 = hn[ctn * 16];
    }
    // Compute tile ct+1 with b1
    update(mmtile(b1), bias1, ((ct + 1) << 4) + l16);
  }

  // Reduce across the 16 lanes holding the same row (masks < 16 stay in-half).
#pragma unroll
  for (int v = 0; v < 8; ++v) {
    float mv = bestv[v];
    int   mi = besti[v];
#pragma unroll
    for (int m = 1; m < 16; m <<= 1) {
      float ov = __shfl_xor(mv, m, 32);
      int   oi = __shfl_xor(mi, m, 32);
      if (ov > mv || (ov == mv && oi < mi)) { mv = ov; mi = oi; }
    }
    if (l16 == 0) {
      int row = row0 + v + 8 * half;
      indices[row] = mi;
      atomicAdd(&hist[mi], 1);
    }
  }
}

// ---------------------------------------------------------------------------
// Kernel 2: quantized gather. q = E[idx]; q_st = x + (q - x) (straight-through
// expression evaluated literally). 128-bit nontemporal stores.
// ---------------------------------------------------------------------------
__global__ __launch_bounds__(256) void vq_gather_kernel(
    const float* __restrict__ x, const float* __restrict__ emb,
    const int* __restrict__ indices, float* __restrict__ q_st,
    float* __restrict__ q_orig, int N) {
  int t = blockIdx.x * blockDim.x + threadIdx.x;  // N * 16 threads (float4 each)
  if (t >= N * 16) return;
  int n = t >> 4, c = t & 15;
  int idx = indices[n];
  v4f q  = *((const v4f*)(emb + (size_t)idx * D_EMB) + c);
  v4f xv = *((const v4f*)(x + (size_t)n * D_EMB) + c);
  v4f st = xv + (q - xv);
  __builtin_nontemporal_store(st, (v4f*)(q_st  + (size_t)n * D_EMB) + c);
  __builtin_nontemporal_store(q,  (v4f*)(q_orig + (size_t)n * D_EMB) + c);
}

// ---------------------------------------------------------------------------
// Kernel 3: one-hot encodings, 536 MB streaming store. enc starts at a
// 4-byte-misaligned offset in d_out, so scalar dwords (fully coalesced + NT).
// ---------------------------------------------------------------------------
__global__ __launch_bounds__(256) void vq_onehot_kernel(
    const int* __restrict__ indices, float* __restrict__ enc,
    long long total, int K) {
  long long t = (long long)blockIdx.x * blockDim.x + threadIdx.x;
  if (t >= total) return;
  int n = (int)(t / K);
  int c = (int)(t - (long long)n * K);
  float v = (indices[n] == c) ? 1.0f : 0.0f;
  __builtin_nontemporal_store(v, enc + t);
}

// ---------------------------------------------------------------------------
// Kernel 4: perplexity = exp(-sum p*log(p+1e-10)), p = hist/N.
// ---------------------------------------------------------------------------
__global__ __launch_bounds__(1024) void vq_perplexity_kernel(
    const int* __restrict__ hist, float* __restrict__ out, int N, int K) {
  __shared__ float sdata[1024];
  int k = threadIdx.x;
  float p = (k < K) ? (float)hist[k] / (float)N : 0.0f;
  sdata[k] = p * logf(p + 1e-10f);
  __syncthreads();
  for (int s = 512; s > 0; s >>= 1) {
    if (k < s) sdata[k] += sdata[k + s];
    __syncthreads();
  }
  if (k == 0) out[0] = expf(-sdata[0]);
}

// ---------------------------------------------------------------------------
extern "C" void kernel_launch(void* const* d_in, const int* in_sizes, int n_in,
                              void* d_out, int out_size, void* d_ws, size_t ws_size,
                              hipStream_t stream) {
  const float* x   = (const float*)d_in[0];   // [32,64,64,64]
  const float* emb = (const float*)d_in[1];   // [1024,64]
  const int N = in_sizes[0] / D_EMB;          // 131072
  const int K = in_sizes[1] / D_EMB;          // 1024

  // d_out layout: q_st [N*64] | q_orig [N*64] | perplexity [1] | encodings [N*K]
  float* q_st   = (float*)d_out;
  float* q_orig = q_st + (size_t)N * D_EMB;
  float* perp   = q_orig + (size_t)N * D_EMB;
  float* enc    = perp + 1;

  // workspace: indices [N] | hist [K] | halfnorm [K]
  int*   indices  = (int*)d_ws;
  int*   hist     = indices + N;
  float* halfnorm = (float*)(hist + K);

  vq_init_kernel<<<(K + 255) / 256, 256, 0, stream>>>(emb, halfnorm, hist, K);

  int waves  = N / 16;                       // one wave per 16-row strip
  int blocks = (waves * 32 + 255) / 256;
  vq_argmin_kernel<<<blocks, 256, 0, stream>>>(x, emb, halfnorm, indices, hist, N, K);

  vq_gather_kernel<<<(N * 16 + 255) / 256, 256, 0, stream>>>(x, emb, indices,
                                                             q_st, q_orig, N);

  long long total = (long long)N * K;
  long long oblocks = (total + 255) / 256;
  vq_onehot_kernel<<<(unsigned)oblocks, 256, 0, stream>>>(indices, enc, total, K);

  vq_perplexity_kernel<<<1, 1024, 0, stream>>>(hist, perp, N, K);
}